// RegionProposalNetwork_4363686773102
// MI455X (gfx1250) — compile-verified
//
#include <hip/hip_runtime.h>
#include <hip/hip_bf16.h>
#include <math.h>

// ---------------- problem constants ----------------
#define NB     8
#define C_IN   512
#define KOUT   512
#define HH     64
#define WW     64
#define MTOT   (NB*HH*WW)        // 32768 spatial positions
#define KTOT   (C_IN*9)          // 4608 reduction length
#define NANCH  (HH*WW*9)         // 36864 anchors per image
#define NPRE   6000
#define NPOST  300
#define MWORDS 192               // 192*32 = 6144 >= 6000 bits per NMS row
#define NEGV   (-1.0e9f)

// ---------------- WMMA types ----------------
typedef __attribute__((ext_vector_type(2))) float v2f;
typedef __attribute__((ext_vector_type(8))) float v8f;

// GEMM tiling: block 128(M) x 128(N), 8 waves of 32(M) x 64(N), K-chunk 32
#define BM2 128
#define BN2 128
#define KCH 32

// =====================================================================
// Stage 0: transpose w1 (ko,c,ky,kx) -> w1t[k][ko] with k = r*512 + c
// =====================================================================
__global__ __launch_bounds__(256) void transpose_w(
    const float* __restrict__ w1, float* __restrict__ w1t)
{
    const int idx = blockIdx.x * 256 + threadIdx.x;   // 0 .. 512*4608-1
    const int ko  = idx & 511;
    const int k   = idx >> 9;          // r*512 + c
    const int r   = k >> 9;
    const int c   = k & 511;
    w1t[idx] = w1[(size_t)ko * KTOT + c * 9 + r];
}

// =====================================================================
// Stage 1: 3x3 conv + bias + ReLU as implicit GEMM on V_WMMA_F32_16X16X4_F32
//   K reordered as (tap r, channel c) -> linear A addresses, no div/mod.
//   A in LDS as [m][k], B in LDS as [n][k]: every WMMA fragment is an
//   aligned contiguous K-pair -> single ds_load_b64 (merge-able 2addr_b64),
//   no register repacking movs.
//   mid stored NHWC: mid[m*512 + ko]
// =====================================================================
__global__ __launch_bounds__(256) void conv3x3_wmma(
    const float* __restrict__ x, const float* __restrict__ w1t,
    const float* __restrict__ b1, float* __restrict__ mid)
{
    __shared__ float As[BM2][KCH + 2];      // 128 x 34 floats
    __shared__ float Bs[BN2][KCH + 2];      // 128 x 34 floats (k-contiguous!)

    const int tid    = threadIdx.x;
    const int mBase  = blockIdx.x * BM2;
    const int koBase = blockIdx.y * BN2;
    const int wv     = tid >> 5;
    const int lane   = tid & 31;
    const int lm     = lane & 15;
    const int hi     = lane >> 4;           // selects K pair within WMMA quad
    const int wvM    = (wv >> 1) * 32;      // wave M sub-tile (4 waves in M)
    const int wvN    = (wv & 1) * 64;       // wave N sub-tile (2 waves in N)

    // staging decomposition: each thread owns a fixed lane (m or n),
    // kLocal = kSub0 + 2*it (16 iterations cover all 32 k's blockwide)
    const int sLane = tid & 127;
    const int kSub0 = tid >> 7;

    // per-thread A position (fixed)
    const int m  = mBase + sLane;
    const int n  = m >> 12;
    const int yx = m & 4095;
    const int yy = yx >> 6;
    const int xx = yx & 63;
    const float* xPlane = x + (size_t)n * C_IN * (HH * WW);

    v8f acc[2][4];
    #pragma unroll
    for (int i = 0; i < 2; ++i)
        #pragma unroll
        for (int j = 0; j < 4; ++j)
            acc[i][j] = (v8f){};

    for (int r = 0; r < 9; ++r) {
        const int ky = r / 3;               // 9x per kernel, negligible
        const int kx = r - ky * 3;
        const int iy = yy + ky - 1;
        const int ix = xx + kx - 1;
        const bool okA = ((unsigned)iy < (unsigned)HH) && ((unsigned)ix < (unsigned)WW);
        const float* aBase = xPlane + iy * WW + ix;          // + c*4096 per k
        const float* bBase = w1t + ((size_t)r * C_IN) * KOUT + koBase + sLane;

        for (int cb = 0; cb < C_IN; cb += KCH) {
            // ---- stage A: 128 x 32, contiguous across m ----
            #pragma unroll
            for (int it = 0; it < 16; ++it) {
                const int kl = kSub0 + 2 * it;
                float v = 0.0f;
                if (okA) v = aBase[(size_t)(cb + kl) * (HH * WW)];
                As[sLane][kl] = v;
            }
            // ---- stage B: coalesced global read, LDS store [n][k] ----
            #pragma unroll
            for (int it = 0; it < 16; ++it) {
                const int kl = kSub0 + 2 * it;
                Bs[sLane][kl] = bBase[(size_t)(cb + kl) * KOUT];
            }
            __syncthreads();

            // ---- 8 K-steps x 8 WMMA (2 A-frags x 4 B-frags) ----
            #pragma unroll
            for (int kk = 0; kk < KCH; kk += 4) {
                v2f a0, a1, b0, b1, b2, b3;
                a0.x = As[wvM + lm][kk + 2 * hi];
                a0.y = As[wvM + lm][kk + 2 * hi + 1];
                a1.x = As[wvM + 16 + lm][kk + 2 * hi];
                a1.y = As[wvM + 16 + lm][kk + 2 * hi + 1];
                b0.x = Bs[wvN + lm][kk + 2 * hi];
                b0.y = Bs[wvN + lm][kk + 2 * hi + 1];
                b1.x = Bs[wvN + 16 + lm][kk + 2 * hi];
                b1.y = Bs[wvN + 16 + lm][kk + 2 * hi + 1];
                b2.x = Bs[wvN + 32 + lm][kk + 2 * hi];
                b2.y = Bs[wvN + 32 + lm][kk + 2 * hi + 1];
                b3.x = Bs[wvN + 48 + lm][kk + 2 * hi];
                b3.y = Bs[wvN + 48 + lm][kk + 2 * hi + 1];
                acc[0][0] = __builtin_amdgcn_wmma_f32_16x16x4_f32(false, a0, false, b0, (short)0, acc[0][0], false, false);
                acc[0][1] = __builtin_amdgcn_wmma_f32_16x16x4_f32(false, a0, false, b1, (short)0, acc[0][1], false, false);
                acc[0][2] = __builtin_amdgcn_wmma_f32_16x16x4_f32(false, a0, false, b2, (short)0, acc[0][2], false, false);
                acc[0][3] = __builtin_amdgcn_wmma_f32_16x16x4_f32(false, a0, false, b3, (short)0, acc[0][3], false, false);
                acc[1][0] = __builtin_amdgcn_wmma_f32_16x16x4_f32(false, a1, false, b0, (short)0, acc[1][0], false, false);
                acc[1][1] = __builtin_amdgcn_wmma_f32_16x16x4_f32(false, a1, false, b1, (short)0, acc[1][1], false, false);
                acc[1][2] = __builtin_amdgcn_wmma_f32_16x16x4_f32(false, a1, false, b2, (short)0, acc[1][2], false, false);
                acc[1][3] = __builtin_amdgcn_wmma_f32_16x16x4_f32(false, a1, false, b3, (short)0, acc[1][3], false, false);
            }
            __syncthreads();
        }
    }

    // ---- epilogue: bias + ReLU, store NHWC ----
    #pragma unroll
    for (int j = 0; j < 4; ++j) {
        const int col = koBase + wvN + 16 * j + lm;
        const float bia = b1[col];
        #pragma unroll
        for (int i = 0; i < 2; ++i) {
            #pragma unroll
            for (int rr = 0; rr < 8; ++rr) {
                const int mrow = mBase + wvM + i * 16 + hi * 8 + rr;
                float v = acc[i][j][rr] + bia;
                v = v > 0.0f ? v : 0.0f;
                mid[(size_t)mrow * KOUT + col] = v;
            }
        }
    }
}

// =====================================================================
// Stage 2: 1x1 heads + softmax + anchor decode + clip + min-size mask
// =====================================================================
__global__ __launch_bounds__(64) void rpn_head(
    const float* __restrict__ mid,
    const float* __restrict__ wsc, const float* __restrict__ bsc,
    const float* __restrict__ wlc, const float* __restrict__ blc,
    float* __restrict__ rpn_locs, float* __restrict__ rpn_scores,
    float* __restrict__ anchorOut, float* __restrict__ fgMasked,
    float* __restrict__ roiAll,
    const int* __restrict__ img_h_p, const int* __restrict__ img_w_p)
{
    __shared__ float row[C_IN];
    __shared__ float outv[54];

    const int m   = blockIdx.x;
    const int tid = threadIdx.x;

    for (int c = tid; c < C_IN; c += 64) row[c] = mid[(size_t)m * C_IN + c];
    __syncthreads();

    for (int o = tid; o < 54; o += 64) {
        const float* wp;
        float s;
        if (o < 18) { wp = wsc + (size_t)o * C_IN;        s = bsc[o]; }
        else        { wp = wlc + (size_t)(o - 18) * C_IN; s = blc[o - 18]; }
        for (int c = 0; c < C_IN; ++c) s += row[c] * wp[c];
        outv[o] = s;
    }
    __syncthreads();

    const int n  = m >> 12;
    const int yx = m & 4095;
    const int yy = yx >> 6;
    const int xx = yx & 63;

    if (tid < 9) {
        const int a = tid;
        const float ratios[3] = {0.5f, 1.0f, 2.0f};
        const float scales[3] = {8.0f, 16.0f, 32.0f};
        const float rr = ratios[a / 3];
        const float ss = scales[a % 3];
        const float ah = 16.0f * ss * sqrtf(rr);
        const float aw = 16.0f * ss * sqrtf(1.0f / rr);
        const float acy = yy * 16.0f + 8.0f;
        const float acx = xx * 16.0f + 8.0f;
        const int aidx = yx * 9 + a;

        if (n == 0) {
            anchorOut[aidx * 4 + 0] = acy - 0.5f * ah;
            anchorOut[aidx * 4 + 1] = acx - 0.5f * aw;
            anchorOut[aidx * 4 + 2] = acy + 0.5f * ah;
            anchorOut[aidx * 4 + 3] = acx + 0.5f * aw;
        }

        const float r0 = outv[2 * a];
        const float r1 = outv[2 * a + 1];
        const float dy = outv[18 + 4 * a + 0];
        const float dx = outv[18 + 4 * a + 1];
        const float dh = outv[18 + 4 * a + 2];
        const float dw = outv[18 + 4 * a + 3];

        size_t sbase = ((size_t)n * NANCH + aidx);
        rpn_scores[sbase * 2 + 0] = r0;
        rpn_scores[sbase * 2 + 1] = r1;
        rpn_locs[sbase * 4 + 0] = dy;
        rpn_locs[sbase * 4 + 1] = dx;
        rpn_locs[sbase * 4 + 2] = dh;
        rpn_locs[sbase * 4 + 3] = dw;

        const float fg = 1.0f / (1.0f + expf(r0 - r1));

        const float fh = (float)(*img_h_p);
        const float fw = (float)(*img_w_p);
        const float cy = dy * ah + acy;
        const float cx = dx * aw + acx;
        const float h  = ah * expf(dh);
        const float w  = aw * expf(dw);
        float y1 = fminf(fmaxf(cy - 0.5f * h, 0.0f), fh);
        float x1 = fminf(fmaxf(cx - 0.5f * w, 0.0f), fw);
        float y2 = fminf(fmaxf(cy + 0.5f * h, 0.0f), fh);
        float x2 = fminf(fmaxf(cx + 0.5f * w, 0.0f), fw);

        roiAll[sbase * 4 + 0] = y1;
        roiAll[sbase * 4 + 1] = x1;
        roiAll[sbase * 4 + 2] = y2;
        roiAll[sbase * 4 + 3] = x2;

        const bool ok = ((y2 - y1) >= 16.0f) && ((x2 - x1) >= 16.0f);
        fgMasked[sbase] = ok ? fg : NEGV;
    }
}

// =====================================================================
// Stage 3: exact descending top-k via padded bitonic sort of (score,idx) keys
// =====================================================================
__global__ __launch_bounds__(256) void sort_init(
    const float* __restrict__ fgMasked, unsigned long long* __restrict__ keys)
{
    const int i = blockIdx.x * 256 + threadIdx.x;     // 0..65535
    const int n = blockIdx.y;
    unsigned long long key = 0ull;                    // padding sinks to bottom
    if (i < NANCH) {
        unsigned u = __float_as_uint(fgMasked[(size_t)n * NANCH + i]);
        u ^= (u & 0x80000000u) ? 0xFFFFFFFFu : 0x80000000u;   // order-preserving flip
        key = ((unsigned long long)u << 32) | (unsigned)(0xFFFFFFFFu - (unsigned)i);
    }
    keys[(size_t)n * 65536 + i] = key;
}

__global__ __launch_bounds__(256) void bitonic_step(
    unsigned long long* __restrict__ keys, int k, int j)
{
    const int i = blockIdx.x * 256 + threadIdx.x;
    unsigned long long* a = keys + (size_t)blockIdx.y * 65536;
    const int l = i ^ j;
    if (l > i) {
        unsigned long long ai = a[i], al = a[l];
        const bool up = (i & k) == 0;                 // descending network
        if (up ? (ai < al) : (ai > al)) { a[i] = al; a[l] = ai; }
    }
}

__global__ __launch_bounds__(256) void gather_top(
    const unsigned long long* __restrict__ keys, const float* __restrict__ roiAll,
    float* __restrict__ topBoxes, int* __restrict__ topValid)
{
    const int t = blockIdx.x * 256 + threadIdx.x;
    const int n = blockIdx.y;
    if (t >= NPRE) return;
    const unsigned long long key = keys[(size_t)n * 65536 + t];
    unsigned idx = 0xFFFFFFFFu - (unsigned)(key & 0xFFFFFFFFu);
    if (idx >= (unsigned)NANCH) idx = 0;
    unsigned u = (unsigned)(key >> 32);
    u = (u & 0x80000000u) ? (u ^ 0x80000000u) : ~u;   // un-flip
    const float score = __uint_as_float(u);
    topValid[(size_t)n * NPRE + t] = (score > NEGV * 0.5f) ? 1 : 0;
    #pragma unroll
    for (int d = 0; d < 4; ++d)
        topBoxes[((size_t)n * NPRE + t) * 4 + d] =
            roiAll[((size_t)n * NANCH + idx) * 4 + d];
}

// =====================================================================
// Stage 4a: IoU suppression bitmask: bit (i, j) set iff j > i and IoU > 0.7
// =====================================================================
__global__ __launch_bounds__(256) void iou_mask(
    const float* __restrict__ topBoxes, unsigned* __restrict__ iouMask)
{
    __shared__ float rb[8][4];
    __shared__ float cb[1024 * 4];

    const int n = blockIdx.y;
    const float* boxes = topBoxes + (size_t)n * NPRE * 4;
    unsigned* maskp    = iouMask + (size_t)n * NPRE * MWORDS;
    const int iBase = blockIdx.x * 8;
    const int tid   = threadIdx.x;

    if (tid < 32) rb[tid >> 2][tid & 3] = boxes[(size_t)(iBase + (tid >> 2)) * 4 + (tid & 3)];

    const int il  = tid >> 5;
    const int wIn = tid & 31;

    for (int chunk = 0; chunk < 6; ++chunk) {
        const int jBase = chunk * 1024;
        __syncthreads();
        for (int e = tid; e < 1024 * 4; e += 256) {
            const int j = jBase + (e >> 2);
            cb[e] = (j < NPRE) ? boxes[(size_t)j * 4 + (e & 3)] : 0.0f;
        }
        __syncthreads();

        const int i = iBase + il;
        const float y1 = rb[il][0], x1 = rb[il][1], y2 = rb[il][2], x2 = rb[il][3];
        const float area_i = (y2 - y1) * (x2 - x1);
        unsigned bits = 0u;
        const int j0 = jBase + wIn * 32;
        #pragma unroll 4
        for (int b = 0; b < 32; ++b) {
            const int j = j0 + b;
            if (j <= i || j >= NPRE) continue;
            const int e = (wIn * 32 + b) * 4;
            const float by1 = cb[e], bx1 = cb[e + 1], by2 = cb[e + 2], bx2 = cb[e + 3];
            float ih = fminf(y2, by2) - fmaxf(y1, by1); ih = ih > 0.0f ? ih : 0.0f;
            float iw = fminf(x2, bx2) - fmaxf(x1, bx1); iw = iw > 0.0f ? iw : 0.0f;
            const float inter = ih * iw;
            const float area_j = (by2 - by1) * (bx2 - bx1);
            const float iou = inter / (area_i + area_j - inter + 1e-9f);
            if (iou > 0.7f) bits |= (1u << b);
        }
        maskp[(size_t)i * MWORDS + chunk * 32 + wIn] = bits;
    }
}

// =====================================================================
// Stage 4b: greedy sequential NMS scan + compaction (one wave32 per image)
// =====================================================================
__global__ __launch_bounds__(32) void nms_scan(
    const float* __restrict__ topBoxes, const int* __restrict__ topValid,
    const unsigned* __restrict__ iouMask,
    float* __restrict__ rois, int* __restrict__ roiIdx)
{
    __shared__ unsigned remS[MWORDS];
    const int n    = blockIdx.x;
    const int lane = threadIdx.x;
    const unsigned* maskp = iouMask + (size_t)n * NPRE * MWORDS;
    const int* validp     = topValid + (size_t)n * NPRE;

    for (int q = lane; q < MWORDS; q += 32) remS[q] = 0u;

    for (int i = 0; i < NPRE; ++i) {
        const unsigned word = remS[i >> 5];              // same-wave LDS: in order
        const bool cur = validp[i] && !((word >> (i & 31)) & 1u);
        if (cur) {
            #pragma unroll
            for (int q = 0; q < 6; ++q)
                remS[lane + 32 * q] |= maskp[(size_t)i * MWORDS + lane + 32 * q];
        }
    }

    for (int t = lane; t < NPOST; t += 32) {
        rois[((size_t)n * NPOST + t) * 4 + 0] = 0.0f;
        rois[((size_t)n * NPOST + t) * 4 + 1] = 0.0f;
        rois[((size_t)n * NPOST + t) * 4 + 2] = 0.0f;
        rois[((size_t)n * NPOST + t) * 4 + 3] = 0.0f;
        roiIdx[(size_t)n * NPOST + t] = -1;
    }

    if (lane == 0) {
        int rank = 0;
        for (int i = 0; i < NPRE && rank < NPOST; ++i) {
            const bool kept = validp[i] && !((remS[i >> 5] >> (i & 31)) & 1u);
            if (kept) {
                #pragma unroll
                for (int d = 0; d < 4; ++d)
                    rois[((size_t)n * NPOST + rank) * 4 + d] =
                        topBoxes[((size_t)n * NPRE + i) * 4 + d];
                roiIdx[(size_t)n * NPOST + rank] = n;
                ++rank;
            }
        }
    }
}

// =====================================================================
// Launch. Output layout (concat, flat, return order):
//   rpn_locs   8*36864*4 = 1179648 @ 0
//   rpn_scores 8*36864*2 =  589824 @ 1179648
//   rois       8*300*4   =    9600 @ 1769472
//   roi_indices(int32) 8*300 = 2400 @ 1779072
//   anchor     36864*4   =  147456 @ 1781472
// =====================================================================
extern "C" void kernel_launch(void* const* d_in, const int* in_sizes, int n_in,
                              void* d_out, int out_size, void* d_ws, size_t ws_size,
                              hipStream_t stream)
{
    const float* x   = (const float*)d_in[0];
    const float* w1  = (const float*)d_in[1];
    const float* b1  = (const float*)d_in[2];
    const float* wsc = (const float*)d_in[3];
    const float* bsc = (const float*)d_in[4];
    const float* wlc = (const float*)d_in[5];
    const float* blc = (const float*)d_in[6];
    const int* img_h = (const int*)d_in[7];
    const int* img_w = (const int*)d_in[8];

    float* out        = (float*)d_out;
    float* rpn_locs   = out;
    float* rpn_scores = out + 1179648;
    float* rois       = out + 1769472;
    int*   roiIdx     = (int*)(out + 1779072);
    float* anchorOut  = out + 1781472;

    char* ws = (char*)d_ws;
    float* mid                 = (float*)(ws);                          // 64 MiB (NHWC)
    float* fgMasked            = (float*)(ws + 67108864);               // 1.125 MiB
    float* roiAll              = (float*)(ws + 68288512);               // 4.5 MiB
    unsigned long long* keys   = (unsigned long long*)(ws + 73007104);  // 4 MiB
    float* topBoxes            = (float*)(ws + 77201408);               // 0.73 MiB
    int*   topValid            = (int*)(ws + 77969408);                 // 0.18 MiB
    unsigned* iouMaskBuf       = (unsigned*)(ws + 78161408);            // 35.2 MiB
    float* w1t                 = (float*)(ws + 115025408);              // 9 MiB -> ~119 MiB total

    // 0) one-time weight transpose for coalesced B staging
    transpose_w<<<dim3((KTOT * KOUT) / 256), 256, 0, stream>>>(w1, w1t);

    // 1) heavy conv on the WMMA pipe
    conv3x3_wmma<<<dim3(MTOT / BM2, KOUT / BN2), 256, 0, stream>>>(x, w1t, b1, mid);

    // 2) 1x1 heads + decode
    rpn_head<<<dim3(MTOT), 64, 0, stream>>>(mid, wsc, bsc, wlc, blc,
                                            rpn_locs, rpn_scores, anchorOut,
                                            fgMasked, roiAll, img_h, img_w);

    // 3) exact ordered top-6000: bitonic sort of 65536 keys per image
    sort_init<<<dim3(256, NB), 256, 0, stream>>>(fgMasked, keys);
    for (int k = 2; k <= 65536; k <<= 1)
        for (int j = k >> 1; j > 0; j >>= 1)
            bitonic_step<<<dim3(256, NB), 256, 0, stream>>>(keys, k, j);
    gather_top<<<dim3((NPRE + 255) / 256, NB), 256, 0, stream>>>(keys, roiAll,
                                                                 topBoxes, topValid);

    // 4) NMS
    iou_mask<<<dim3(NPRE / 8, NB), 256, 0, stream>>>(topBoxes, iouMaskBuf);
    nms_scan<<<dim3(NB), 32, 0, stream>>>(topBoxes, topValid, iouMaskBuf, rois, roiIdx);
}